// PIMLHBV_Static_8675833938686
// MI455X (gfx1250) — compile-verified
//
#include <hip/hip_runtime.h>

#define BDIM 2048
#define TDIM 4096

typedef float v2f __attribute__((ext_vector_type(2)));
typedef float v8f __attribute__((ext_vector_type(8)));

// heaviside(y) = (tanh(5y)+1)/2 == sigmoid(10y) == 1/(1 + exp2(-10*log2(e)*y))
__device__ __forceinline__ float sigm10(float y) {
    float e = __builtin_amdgcn_exp2f(-14.4269504088896341f * y);  // v_exp_f32
    return __builtin_amdgcn_rcpf(1.0f + e);                        // v_rcp_f32
}

__device__ __forceinline__ float clip1e5(float d) {
    return fminf(fmaxf(d, -100000.0f), 100000.0f);
}

// ---------------------------------------------------------------------------
// Phase 1: sequential HBV recurrence. One thread per basin (2048 threads).
// Latency-bound; 7 sigmoids + 1 pow per step (exploiting heaviside(-x)=1-h(x)).
// q written time-major (q[t*B+b]) -> coalesced stores, sets up WMMA phase.
// ---------------------------------------------------------------------------
__global__ __launch_bounds__(256) void hbv_scan_kernel(
    const float* __restrict__ x, const float* __restrict__ p,
    float* __restrict__ qws)
{
    const int b = blockIdx.x * 256 + threadIdx.x;   // grid exactly covers BDIM

    const float Tt     = p[0]  * -1.5f;
    const float CFR    = p[1]  * 0.01f;
    const float CFMAX  = p[2];
    const float SCF    = p[3];
    const float FC     = p[4]  * 100.0f;
    const float Beta   = p[5]  * 8.0f;
    const float LP     = p[6]  * 2.5f;
    const float CWH    = p[7]  * 0.3f;
    const float UZL    = p[8]  * 120.0f;
    const float K0     = p[9]  * 0.5f;
    const float K1     = p[10] * 0.3f;
    const float K2     = p[11] * 0.01f;
    const float PER    = p[12] * 3.5f;
    const float rawCWH = p[7];

    const float invFC   = __builtin_amdgcn_rcpf(FC);
    const float FCLP    = FC * LP;
    const float invFCLP = __builtin_amdgcn_rcpf(FCLP);
    const float CFRMAX  = CFR * CFMAX;

    float S1 = 0.0f, S2 = 0.0f, S3 = 0.0f, S4 = 0.0f, S5 = 0.0f;

    const float* xb = x + (size_t)b * TDIM * 3;
    float* qp = qws + b;

    for (int t = 0; t < TDIM; ++t) {
        float P   = xb[0];
        float Tm  = xb[1];
        float PET = xb[2];
        xb += 3;

        float h_cold = sigm10(Tt - Tm);
        float h_warm = 1.0f - h_cold;               // heaviside(T-Tt) = 1 - heaviside(Tt-T)
        float Ps  = P * SCF * h_cold;
        float Rfr = fminf((Tt - Tm) * CFRMAX, S2) * h_cold;
        float Pr  = P * h_warm;
        float M   = fminf((Tm - Tt) * CFMAX, S1) * h_warm;
        float I   = sigm10(S2 - S1 * CWH) * (Pr + M);
        // (S3/FC)^Beta via native log2/exp2 (0 -> -inf -> 0, matches pow(0,Beta>0))
        float Re  = I * __builtin_amdgcn_exp2f(Beta * __builtin_amdgcn_logf(S3 * invFC));
        float hS3 = sigm10(S3);
        float hLP = sigm10(FCLP - S3);
        float Ea  = hS3 * (PET * S3 * invFCLP) * hLP + hS3 * PET * (1.0f - hLP);
        float hS4 = sigm10(S4);
        float Pe  = PER * hS4;
        float q0  = (S4 - UZL) * K0 * sigm10(S4 - UZL);
        float q1  = S4 * K1 * hS4;
        float q2  = S5 * K2 * sigm10(S5);

        float nS1 = fmaxf(S1 + clip1e5(Ps + Rfr - M), 0.0f);
        float nS2 = fmaxf(fminf(S2 + clip1e5(Pr + M - I - Rfr), nS1 * rawCWH), 0.0f);
        float nS3 = fmaxf(S3 + clip1e5(I - Ea - Re), 0.0f);
        float nS4 = fmaxf(S4 + clip1e5(Re - q0 - q1 - Pe), 0.0f);
        float nS5 = fmaxf(S5 + clip1e5(Pe - q2), 0.0f);
        S1 = nS1; S2 = nS2; S3 = nS3; S4 = nS4; S5 = nS5;

        qp[(size_t)t * BDIM] = q0 + q1 + q2;        // coalesced across the wave
    }
}

// ---------------------------------------------------------------------------
// Phase 2: 9-tap SAME conv along time as a banded matmul on the matrix core.
// Out(16 basins x 16 times) = Q(16x24) x Wband(24x16), K split into 6 WMMAs.
// One wave per 16x16 tile; 32768 tiles; EXEC all-1s (exact grid, no early exit,
// branchless clamped loads -> no v_cmpx/s_cbranch, loads pipeline together).
// ---------------------------------------------------------------------------
__global__ __launch_bounds__(256) void hbv_route_wmma_kernel(
    const float* __restrict__ qws, float* __restrict__ out)
{
    // Reference weight constants, folded to f32 at compile time (INV_MID is 0.25f,
    // INV_WSUM 1 ulp below 0.25f; ~6e-8 relative weight error, below f32 conv roundoff).
    constexpr double INV_MID_D = 1.0 / (4.0 + 1e-7);
    constexpr double WSUM_D = 1.0 + 2.0 * ((1.0 - 1.0 * INV_MID_D) +
                                           (1.0 - 2.0 * INV_MID_D) +
                                           (1.0 - 3.0 * INV_MID_D) +
                                           (1.0 - 4.0 * INV_MID_D));
    constexpr float INV_MID  = (float)INV_MID_D;
    constexpr float INV_WSUM = (float)(1.0 / (WSUM_D + 1e-7));

    const int lane = threadIdx.x & 31;
    const int wave = threadIdx.x >> 5;
    const int tile = blockIdx.x * 8 + wave;     // (BDIM/16)*(TDIM/16) tiles
    const int tb   = tile & (BDIM / 16 - 1);    // basin tile (128)
    const int tt   = tile >> 7;                 // time tile (256)
    const int b0   = tb * 16;
    const int t0   = tt * 16;
    const int mn   = lane & 15;                 // A: basin row / B,D: time col
    const int khi  = lane >> 4;                 // K half-select per ISA VGPR layout

    // Gather all 12 A fragments (unconditional clamped loads, zeroed by select)
    // and all 12 weight fragments before the WMMA chain.
    v2f afr[6], wfr[6];
#pragma unroll
    for (int c = 0; c < 6; ++c) {
#pragma unroll
        for (int j = 0; j < 2; ++j) {
            const int kg = 4 * c + j + 2 * khi;        // global K index in [0,24)
            // A[m=basin, kg] = q[b0+m, t0-4+kg], zero-padded in time
            const int tq  = t0 - 4 + kg;
            const int tqc = tq < 0 ? 0 : (tq > TDIM - 1 ? TDIM - 1 : tq);
            const float v = qws[(size_t)tqc * BDIM + b0 + mn];
            afr[c][j] = (tq == tqc) ? v : 0.0f;
            // Wband[kg, n] = w[kg-n]; triangular formula is 0 outside band
            const float dd = fabsf((float)(kg - mn) - 4.0f);
            wfr[c][j] = fmaxf(0.0f, fmaf(-dd, INV_MID, 1.0f)) * INV_WSUM;
        }
    }

    v8f acc = {0.0f, 0.0f, 0.0f, 0.0f, 0.0f, 0.0f, 0.0f, 0.0f};
#pragma unroll
    for (int c = 0; c < 6; ++c) {
        // D = A(16x4) * B(4x16) + C   -> v_wmma_f32_16x16x4_f32
        acc = __builtin_amdgcn_wmma_f32_16x16x4_f32(
            false, afr[c], false, wfr[c], (short)0, acc, false, false);
    }

    // D layout: VGPR v, lanes 0-15 -> M=v, lanes 16-31 -> M=v+8; N = lane&15.
    // M = basin, N = time -> each half-wave stores a 64B contiguous run.
#pragma unroll
    for (int v = 0; v < 8; ++v) {
        const int m = v + 8 * khi;
        out[(size_t)(b0 + m) * TDIM + t0 + mn] = acc[v];
    }
}

extern "C" void kernel_launch(void* const* d_in, const int* in_sizes, int n_in,
                              void* d_out, int out_size, void* d_ws, size_t ws_size,
                              hipStream_t stream) {
    (void)in_sizes; (void)n_in; (void)out_size; (void)ws_size;
    const float* x      = (const float*)d_in[0];   // (B, T, 3) f32
    const float* params = (const float*)d_in[1];   // (13,) f32
    float* out = (float*)d_out;                    // (B, T, 1) f32
    float* qws = (float*)d_ws;                     // B*T f32 scratch, time-major

    hbv_scan_kernel<<<BDIM / 256, 256, 0, stream>>>(x, params, qws);
    hbv_route_wmma_kernel<<<(BDIM / 16) * (TDIM / 16) / 8, 256, 0, stream>>>(qws, out);
}